// CTCLossModel_13065290514981
// MI455X (gfx1250) — compile-verified
//
#include <hip/hip_runtime.h>
#include <hip/hip_bf16.h>

// ---------------------------------------------------------------------------
// CTC loss, forward only.  B=64, T=2000, C=128 (blank=127), L=200 -> S=401.
// Kernel 1: per-(b,t) row logsumexp  (memory-bound streaming, wave-per-row)
// Kernel 2: per-b alpha scan, one workgroup (16 waves) per batch element.
//           gfx1250 path: GLOBAL_LOAD_ASYNC_TO_LDS_B128 ring prefetch
//           (ASYNCcnt) + split s_barrier_signal/-wait per time step.
// ---------------------------------------------------------------------------

#define NEGF (-1e30f)

constexpr int CC        = 128;   // classes incl. blank
constexpr int BLANK     = 127;
constexpr int S_MAX     = 401;   // 2*L+1
constexpr int NBUF      = 8;     // row ring slots (power of two)
constexpr int LOOKAHEAD = 6;     // rows in flight beyond current
constexpr int SCAN_BLK  = 512;   // 16 waves

// gfx1250 async DMA: each lane copies 16 bytes from the global address in vaddr to the LDS address in vdst; tracked by ASYNCcnt.
#define ASYNC_B128(lds_off_u32, gptr)                                          \
  asm volatile("global_load_async_to_lds_b128 %0, %1, off"                     \
               :: "v"(lds_off_u32), "v"(gptr) : "memory")

#define S_WAIT_ASYNC(n) asm volatile("s_wait_asynccnt " #n ::: "memory")

// Split barrier: LDS stores must land (dscnt) before we signal.
#define BAR_SIGNAL() asm volatile("s_wait_dscnt 0x0\n\ts_barrier_signal -1" ::: "memory")
#define BAR_WAIT()   asm volatile("s_barrier_wait -1" ::: "memory")

// Flat address of a __shared__ object is {LDS_aperture_hi32, lds_offset32};
// truncation to 32 bits yields the LDS byte offset the async op wants.
__device__ __forceinline__ unsigned lds_off_of(const void* p) {
  return (unsigned)(unsigned long long)p;
}

// ---------------------------------------------------------------------------
// Kernel 1: lse[b*T+t] = logsumexp_c y_pred[b,t,c]   (one wave per row)
// ---------------------------------------------------------------------------
__global__ __launch_bounds__(256) void ctc_row_lse(const float* __restrict__ y,
                                                   float* __restrict__ rowlse,
                                                   int nrows) {
  const int lane = threadIdx.x & 31;
  const int row  = blockIdx.x * (blockDim.x >> 5) + (threadIdx.x >> 5);
  if (row >= nrows) return;

  const float4 v = ((const float4*)(y + (size_t)row * CC))[lane];

  float m = fmaxf(fmaxf(v.x, v.y), fmaxf(v.z, v.w));
  #pragma unroll
  for (int off = 16; off > 0; off >>= 1) m = fmaxf(m, __shfl_xor(m, off, 32));

  float s = __expf(v.x - m) + __expf(v.y - m) + __expf(v.z - m) + __expf(v.w - m);
  #pragma unroll
  for (int off = 16; off > 0; off >>= 1) s += __shfl_xor(s, off, 32);

  if (lane == 0) rowlse[row] = m + __logf(s);
}

// ---------------------------------------------------------------------------
// Kernel 2: alpha recursion, one workgroup (16 waves) per batch element.
// ---------------------------------------------------------------------------
__global__ __launch_bounds__(SCAN_BLK) void ctc_scan(
    const float* __restrict__ yp,   const int* __restrict__ yt,
    const int*   __restrict__ ilen, const int* __restrict__ llen,
    const float* __restrict__ rowlse, float* __restrict__ out,
    int T, int L) {
  __shared__ float rowbuf[NBUF][CC];        // async DMA ring (4 KB)
  __shared__ float alphaBuf[2][S_MAX + 2];  // [0..1] = NEG pads, state s at s+2

  const int tid  = threadIdx.x;
  const int lane = tid & 31;
  const int wave = tid >> 5;
  const int b    = blockIdx.x;
  const int S    = 2 * L + 1;

  const float* Y   = yp     + (size_t)b * T * CC;
  const float* LSE = rowlse + (size_t)b * T;

  // Per-thread lattice state (state s == thread tid): class + skip flag.
  int  myc  = BLANK;
  bool mysk = false;
  if (tid < S) {
    if (tid & 1) {
      const int j = (tid - 1) >> 1;
      int l = yt[b * L + j]; if (l < 0) l = 0;
      myc = l;
      if (tid == 1) mysk = true;
      else { int lp = yt[b * L + j - 1]; if (lp < 0) lp = 0; mysk = (l != lp); }
    }
    alphaBuf[0][tid + 2] = NEGF;
    alphaBuf[1][tid + 2] = NEGF;
  }
  if (tid < 2) { alphaBuf[0][tid] = NEGF; alphaBuf[1][tid] = NEGF; }

  unsigned rb_off[NBUF];
  #pragma unroll
  for (int i = 0; i < NBUF; ++i) rb_off[i] = lds_off_of(&rowbuf[i][0]);

  // Each wave DMAs a disjoint 32B slice of each row: lanes 0..1, b128 each.
  const unsigned slice_lds = (unsigned)(wave * 32 + lane * 16);
  const int      slice_elt = wave * 8 + lane * 4;

  // Prologue: rows 0..LOOKAHEAD in flight, then drain so row 0 is ready.
  #pragma unroll
  for (int r = 0; r <= LOOKAHEAD; ++r) {
    if (r < T && lane < 2)
      ASYNC_B128(rb_off[r] + slice_lds, Y + (size_t)r * CC + slice_elt);
  }
  S_WAIT_ASYNC(0);
  __syncthreads();   // all slices of row 0 + alpha init visible block-wide

  // t = 0 initialization.
  if (tid == 0) alphaBuf[0][2] = rowbuf[0][BLANK] - LSE[0];
  if (tid == 1) alphaBuf[0][3] = rowbuf[0][myc] - LSE[0];

  const int len = ilen[b];

  for (int t = 1; t < T; ++t) {
    const int pr = t + LOOKAHEAD;
    const float lse_t = LSE[t];          // issued pre-barrier: latency hidden
    if (pr < T) {
      if (lane < 2)
        ASYNC_B128(rb_off[pr & (NBUF - 1)] + slice_lds,
                   Y + (size_t)pr * CC + slice_elt);
      S_WAIT_ASYNC(6);   // <=6 outstanding => this wave's slice of row t done
    } else {
      S_WAIT_ASYNC(0);   // tail: drain everything
    }
    BAR_SIGNAL();        // alpha(t-1) stores + row-t slice complete first
    BAR_WAIT();          // all waves arrived: full row t resident in LDS

    if (tid < S) {
      const float* row = rowbuf[t & (NBUF - 1)];
      const float* ap  = alphaBuf[(t - 1) & 1];
      const float a0 = ap[tid + 2];
      const float a1 = ap[tid + 1];
      const float a2 = mysk ? ap[tid] : NEGF;
      const float m  = fmaxf(a0, fmaxf(a1, a2));
      const float v  = m + __logf(__expf(a0 - m) + __expf(a1 - m) + __expf(a2 - m))
                         + (row[myc] - lse_t);
      alphaBuf[t & 1][tid + 2] = (t < len) ? v : a0;   // freeze past seq end
    }
  }

  __syncthreads();
  if (tid == 0) {
    const float* af = alphaBuf[(T - 1) & 1];
    const int    se = 2 * llen[b];
    const float  aL = af[se + 2], aP = af[se + 1];
    const float  m  = fmaxf(aL, aP);
    out[b] = -(m + __logf(__expf(aL - m) + __expf(aP - m)));
  }
}

// ---------------------------------------------------------------------------
extern "C" void kernel_launch(void* const* d_in, const int* in_sizes, int n_in,
                              void* d_out, int out_size, void* d_ws, size_t ws_size,
                              hipStream_t stream) {
  const float* y_pred        = (const float*)d_in[0];
  const int*   y_true        = (const int*)  d_in[1];
  const int*   input_lengths = (const int*)  d_in[2];
  const int*   label_lengths = (const int*)  d_in[3];
  float*       out           = (float*)d_out;

  const int B = in_sizes[2];                 // 64
  const int L = in_sizes[1] / B;             // 200
  const int T = in_sizes[0] / (B * CC);      // 2000

  float* rowlse = (float*)d_ws;              // B*T floats (512 KB)
  const int nrows = B * T;

  ctc_row_lse<<<(nrows + 7) / 8, 256, 0, stream>>>(y_pred, rowlse, nrows);
  ctc_scan<<<B, SCAN_BLK, 0, stream>>>(y_pred, y_true, input_lengths,
                                       label_lengths, rowlse, out, T, L);
}